// CoAttention_25486335934535
// MI455X (gfx1250) — compile-verified
//
#include <hip/hip_runtime.h>
#include <hip/hip_bf16.h>
#include <math.h>

#define EPSF 1e-8f

typedef float v2f __attribute__((ext_vector_type(2)));
typedef float v8f __attribute__((ext_vector_type(8)));

__device__ __forceinline__ float wave_sum(float v) {
    for (int o = 16; o > 0; o >>= 1) v += __shfl_xor(v, o, 32);
    return v;
}
__device__ __forceinline__ float wave_max(float v) {
    for (int o = 16; o > 0; o >>= 1) v = fmaxf(v, __shfl_xor(v, o, 32));
    return v;
}

// ---------------------------------------------------------------------------
// Pad raw inputs (B,512,101) -> (B,512,104), optionally negating time column
// ---------------------------------------------------------------------------
__global__ void pad_x_kernel(const float* __restrict__ X, float* __restrict__ Xp, int flip) {
    int t = threadIdx.x;                 // 0..127
    if (t >= 104) return;
    size_t row = blockIdx.x, b = blockIdx.y;
    float v = 0.f;
    if (t < 101) v = X[(b * 512 + row) * 101 + t];
    if (flip && t == 0) v = -v;
    Xp[(b * 512 + row) * 104 + t] = v;
}

// Pad weight (O,101) -> (Opad,104) and bias (O) -> (Opad), zeros elsewhere
__global__ void pad_w_kernel(const float* __restrict__ W, const float* __restrict__ bias,
                             float* __restrict__ Wp, float* __restrict__ bp, int O) {
    int n = blockIdx.x;                  // 0..Opad-1
    int t = threadIdx.x;                 // 0..127
    if (t < 104) {
        float v = (n < O && t < 101) ? W[(size_t)n * 101 + t] : 0.f;
        Wp[(size_t)n * 104 + t] = v;
    }
    if (t == 0) bp[n] = (n < O) ? bias[n] : 0.f;
}

// ---------------------------------------------------------------------------
// Register-blocked WMMA fp32 GEMM (A @ B^T):
//   Out[m,n] = alpha * sum_k A[m,k]*Brows[n,k] + (bias ? bias[n] : beta)
// One wave computes a 16 x (16*NT) macro-tile: 1 A fragment reused across NT
// B fragments / NT accumulators. M = 512 fixed, K multiple of 4, padded bufs.
// ---------------------------------------------------------------------------
template <int NT>
__global__ void gemm_abt_nt(const float* __restrict__ A, int lda, size_t strA,
                            const float* __restrict__ Br, int ldb, size_t strB,
                            float* __restrict__ Out, int ldo, size_t strO,
                            const float* __restrict__ bias, float alpha, float beta,
                            int nMacro, int K) {
    int wv = threadIdx.x >> 5, ln = threadIdx.x & 31;
    int tile = blockIdx.x * 8 + wv;
    if (tile >= 32 * nMacro) return;
    int tm = tile / nMacro, tb = tile % nMacro;
    const float* Ab = A + strA * blockIdx.y;
    const float* Bb = Br + strB * blockIdx.y;
    float* Ob = Out + strO * blockIdx.y;
    int hl = ln & 15;
    int kh = (ln >> 4) << 1;              // 0 or 2
    const float* arow = Ab + (size_t)(tm * 16 + hl) * lda + kh;
    const float* brow[NT];
#pragma unroll
    for (int j = 0; j < NT; ++j)
        brow[j] = Bb + (size_t)((tb * NT + j) * 16 + hl) * ldb + kh;
    v8f zero = {0.f, 0.f, 0.f, 0.f, 0.f, 0.f, 0.f, 0.f};
    v8f acc[NT];
#pragma unroll
    for (int j = 0; j < NT; ++j) acc[j] = zero;
    for (int k = 0; k < K; k += 4) {
        v2f a;
        a.x = arow[k]; a.y = arow[k + 1];
        // locality=3 -> WGP-scope prefetch (pulls into all cache levels)
        if (k + 64 < K) __builtin_prefetch(arow + k + 64, 0, 3);
#pragma unroll
        for (int j = 0; j < NT; ++j) {
            v2f b;
            b.x = brow[j][k]; b.y = brow[j][k + 1];
            acc[j] = __builtin_amdgcn_wmma_f32_16x16x4_f32(false, a, false, b,
                                                           (short)0, acc[j], false, false);
        }
    }
    int row0 = tm * 16 + ((ln >> 4) << 3);
#pragma unroll
    for (int j = 0; j < NT; ++j) {
        int col = (tb * NT + j) * 16 + hl;
        float addv = bias ? bias[col] : beta;
#pragma unroll
        for (int v = 0; v < 8; ++v)
            Ob[(size_t)(row0 + v) * ldo + col] = alpha * acc[j][v] + addv;
    }
}

// ---------------------------------------------------------------------------
// Register-blocked WMMA fp32 GEMM (A @ B), B read column-wise (non-transposed):
//   Out[m,n] = sum_k A[m,k] * V[k,n]
// Used for the (already softmax-normalized) attention @ value products.
// ---------------------------------------------------------------------------
template <int NT>
__global__ void gemm_ab_nt(const float* __restrict__ A, int lda, size_t strA,
                           const float* __restrict__ V, int ldv, size_t strV,
                           float* __restrict__ Out, int ldo, size_t strO,
                           int nMacro, int K) {
    int wv = threadIdx.x >> 5, ln = threadIdx.x & 31;
    int tile = blockIdx.x * 8 + wv;
    if (tile >= 32 * nMacro) return;
    int tm = tile / nMacro, tb = tile % nMacro;
    const float* Ab = A + strA * blockIdx.y;
    const float* Vb = V + strV * blockIdx.y;
    float* Ob = Out + strO * blockIdx.y;
    int hl = ln & 15;
    int kh = (ln >> 4) << 1;
    const float* arow = Ab + (size_t)(tm * 16 + hl) * lda + kh;
    int cols[NT];
#pragma unroll
    for (int j = 0; j < NT; ++j) cols[j] = (tb * NT + j) * 16 + hl;
    v8f zero = {0.f, 0.f, 0.f, 0.f, 0.f, 0.f, 0.f, 0.f};
    v8f acc[NT];
#pragma unroll
    for (int j = 0; j < NT; ++j) acc[j] = zero;
    for (int k0 = 0; k0 < K; k0 += 4) {
        int k = k0 + kh;
        v2f a;
        a.x = arow[k0]; a.y = arow[k0 + 1];
        // locality=3 -> WGP-scope prefetch (pulls into all cache levels)
        if (k0 + 64 < K) __builtin_prefetch(arow + k0 + 64, 0, 3);
#pragma unroll
        for (int j = 0; j < NT; ++j) {
            v2f b;
            b.x = Vb[(size_t)k * ldv + cols[j]];
            b.y = Vb[(size_t)(k + 1) * ldv + cols[j]];
            acc[j] = __builtin_amdgcn_wmma_f32_16x16x4_f32(false, a, false, b,
                                                           (short)0, acc[j], false, false);
        }
    }
    int row0 = tm * 16 + ((ln >> 4) << 3);
#pragma unroll
    for (int j = 0; j < NT; ++j) {
#pragma unroll
        for (int v = 0; v < 8; ++v)
            Ob[(size_t)(row0 + v) * ldo + cols[j]] = acc[j][v];
    }
}

// ---------------------------------------------------------------------------
// In-place Lorentz-linear fixup per row: time = sigmoid(h0)*exp(ls)+1.1;
// space *= sqrt((time^2-1)/clip(sum space^2)). One wave per row.
// ---------------------------------------------------------------------------
__global__ void lorentz_fix_kernel(float* __restrict__ H, int ld, int O,
                                   const float* __restrict__ ls) {
    int wv = threadIdx.x >> 5, ln = threadIdx.x & 31;
    int row = blockIdx.x * 8 + wv;
    float* h = H + ((size_t)blockIdx.y * 512 + row) * ld;
    float sc = expf(ls[0]);
    float h0 = h[0];
    float time = (1.f / (1.f + expf(-h0))) * sc + 1.1f;
    float vals[4]; int idx[4];
    float ssq = 0.f;
    for (int t = 0; t < 4; ++t) {
        int i = 1 + ln + 32 * t;
        idx[t] = i;
        float v = (i < O) ? h[i] : 0.f;
        vals[t] = v;
        ssq += v * v;
    }
    ssq = wave_sum(ssq);
    float sq = sqrtf((time * time - 1.f) / fmaxf(ssq, EPSF));
    if (ln == 0) h[0] = time;
    for (int t = 0; t < 4; ++t)
        if (idx[t] < O) h[idx[t]] = vals[t] * sq;
}

// ---------------------------------------------------------------------------
// Row/col softmax stats of C (512x512 per batch). z=0: rows, z=1: cols.
// ---------------------------------------------------------------------------
__global__ void stats_kernel(const float* __restrict__ Cm,
                             float* __restrict__ rmax, float* __restrict__ rsum,
                             float* __restrict__ cmax, float* __restrict__ csum) {
    int wv = threadIdx.x >> 5, ln = threadIdx.x & 31;
    int idx = blockIdx.x * 8 + wv;
    const float* Cb = Cm + (size_t)blockIdx.y * 512 * 512;
    float vb[16];
    if (blockIdx.z == 0) {
        for (int t = 0; t < 16; ++t) vb[t] = Cb[(size_t)idx * 512 + ln + 32 * t];
    } else {
        for (int t = 0; t < 16; ++t) vb[t] = Cb[(size_t)(ln + 32 * t) * 512 + idx];
    }
    float m = -1e30f;
    for (int t = 0; t < 16; ++t) m = fmaxf(m, vb[t]);
    m = wave_max(m);
    float s = 0.f;
    for (int t = 0; t < 16; ++t) s += expf(vb[t] - m);
    s = wave_sum(s);
    if (ln == 0) {
        size_t o = (size_t)blockIdx.y * 512 + idx;
        if (blockIdx.z == 0) { rmax[o] = m; rsum[o] = s; }
        else                 { cmax[o] = m; csum[o] = s; }
    }
}

// ---------------------------------------------------------------------------
// One pass over C: overwrite C in place with row-softmax A_cs, and write the
// column-softmax A_sc *transposed* (via LDS tile) so both centroid GEMMs get
// contiguous, exp-free A fragments.
// ---------------------------------------------------------------------------
__global__ void normexp_kernel(float* __restrict__ Cm,
                               const float* __restrict__ rmax, const float* __restrict__ rsum,
                               const float* __restrict__ cmax, const float* __restrict__ csum,
                               float* __restrict__ EscT) {
    __shared__ float tle[32][33];
    int lx = threadIdx.x;                // 0..31
    int ly = threadIdx.y;                // 0..7
    size_t b = blockIdx.z;
    int r0 = blockIdx.y * 32, c0 = blockIdx.x * 32;
    float* Cb = Cm + b * (size_t)(512 * 512);
    float* Eb = EscT + b * (size_t)(512 * 512);
    const float* rm = rmax + b * 512; const float* rs = rsum + b * 512;
    const float* cm = cmax + b * 512; const float* cs = csum + b * 512;
    int c = c0 + lx;
    float cmv = cm[c], csv = cs[c];
#pragma unroll
    for (int q = 0; q < 4; ++q) {
        int lr = ly + 8 * q;
        int r = r0 + lr;
        float v = Cb[(size_t)r * 512 + c];
        Cb[(size_t)r * 512 + c] = expf(v - rm[r]) / rs[r];   // A_cs in place
        tle[lr][lx] = expf(v - cmv) / csv;                   // A_sc, staged
    }
    __syncthreads();
#pragma unroll
    for (int q = 0; q < 4; ++q) {
        int lr = ly + 8 * q;
        int j = c0 + lr;                 // output row (sentence index)
        int i = r0 + lx;                 // output col (comment index)
        Eb[(size_t)j * 512 + i] = tle[lx][lr];
    }
}

// ---------------------------------------------------------------------------
// Per-row: centroid normalize -> logmap0 -> transp0 -> expmap -> tanh/project
// -> attention score vs w_h. One wave per row; 129-dim rows (ld=144).
// ---------------------------------------------------------------------------
__global__ void hs_score_kernel(const float* __restrict__ Cent, const float* __restrict__ X,
                                const float* __restrict__ wh, float* __restrict__ sc_out) {
    int wv = threadIdx.x >> 5, ln = threadIdx.x & 31;
    int row = blockIdx.x * 8 + wv;
    size_t base = ((size_t)blockIdx.y * 512 + row) * 144;
    const float* cr = Cent + base;
    const float* xr = X + base;
    float a0 = cr[0], x0 = xr[0];
    float asp[4], xsp[4], wsp[4];
    float ss = 0.f;
    for (int t = 0; t < 4; ++t) {
        int i = 1 + ln + 32 * t;
        asp[t] = cr[i]; xsp[t] = xr[i]; wsp[t] = wh[i];
        ss += asp[t] * asp[t];
    }
    ss = wave_sum(ss);
    float denom = sqrtf(fmaxf(a0 * a0 - ss, EPSF));
    float y0 = a0 / denom;
    float y0c = fmaxf(y0, 1.f + 1e-7f);
    float t2 = y0c * y0c - 1.f;
    float coef = logf(y0c + sqrtf(t2)) / sqrtf(fmaxf(t2, EPSF));
    float vsp[4];
    float dxv = 0.f;
    for (int t = 0; t < 4; ++t) {
        vsp[t] = coef * (asp[t] / denom);
        dxv += xsp[t] * vsp[t];
    }
    dxv = wave_sum(dxv);
    float c2 = dxv / (1.f + x0);
    float u0 = c2 * (x0 + 1.f);
    float usp[4];
    float uss = 0.f;
    for (int t = 0; t < 4; ++t) {
        usp[t] = vsp[t] + c2 * xsp[t];
        uss += usp[t] * usp[t];
    }
    uss = wave_sum(uss);
    float un = sqrtf(fmaxf(uss - u0 * u0, EPSF));
    float eu = expf(un), iu = 1.f / eu;
    float ch = 0.5f * (eu + iu);
    float sh = 0.5f * (eu - iu) / un;
    float st = 0.f, dwt = 0.f;
    for (int t = 0; t < 4; ++t) {
        float z = ch * xsp[t] + sh * usp[t];
        float th = tanhf(z);
        st += th * th;
        dwt += wsp[t] * th;
    }
    st = wave_sum(st);
    dwt = wave_sum(dwt);
    float timeH = sqrtf(1.f + st);
    float score = 2.f + 2.f * (-wh[0] * timeH + dwt);
    if (ln == 0) sc_out[(size_t)blockIdx.y * 512 + row] = score;
}

// ---------------------------------------------------------------------------
// Softmax over 512 scores per batch; y=0 -> a_s, y=1 -> a_c. One wave.
// ---------------------------------------------------------------------------
__global__ void softmax512_kernel(const float* __restrict__ scs, const float* __restrict__ scc,
                                  float* __restrict__ outs, float* __restrict__ outc) {
    int ln = threadIdx.x;
    size_t b = blockIdx.x;
    const float* s = (blockIdx.y ? scc : scs) + b * 512;
    float* o = (blockIdx.y ? outc : outs) + b * 512;
    float vb[16];
    float m = -1e30f;
    for (int t = 0; t < 16; ++t) { vb[t] = s[ln + 32 * t]; m = fmaxf(m, vb[t]); }
    m = wave_max(m);
    float sum = 0.f;
    for (int t = 0; t < 16; ++t) { vb[t] = expf(vb[t] - m); sum += vb[t]; }
    sum = wave_sum(sum);
    float r = 1.f / sum;
    for (int t = 0; t < 16; ++t) o[ln + 32 * t] = vb[t] * r;
}

// ---------------------------------------------------------------------------
// Final: s = centroid(Xs, a_s), c = centroid(Xc, a_c), co = project(concat)
// ---------------------------------------------------------------------------
__global__ void final_kernel(const float* __restrict__ as_, const float* __restrict__ ac_,
                             const float* __restrict__ Xs, const float* __restrict__ Xc,
                             float* __restrict__ co) {
    __shared__ float s_avg[101], c_avg[101], scal[3];
    int t = threadIdx.x;
    size_t b = blockIdx.x;
    const float* a_s = as_ + b * 512;
    const float* a_c = ac_ + b * 512;
    if (t < 101) {
        const float* X = Xs + b * (size_t)512 * 101;
        float acc = 0.f;
        for (int i = 0; i < 512; ++i) acc += a_s[i] * X[(size_t)i * 101 + t];
        s_avg[t] = acc;
    } else if (t >= 128 && t < 229) {
        int k = t - 128;
        const float* X = Xc + b * (size_t)512 * 101;
        float acc = 0.f;
        for (int i = 0; i < 512; ++i) acc += a_c[i] * X[(size_t)i * 101 + k];
        c_avg[k] = acc;
    }
    __syncthreads();
    if (t < 32) {
        float ss = 0.f, cc = 0.f;
        for (int q = 1 + t; q <= 100; q += 32) {
            ss += s_avg[q] * s_avg[q];
            cc += c_avg[q] * c_avg[q];
        }
        ss = wave_sum(ss);
        cc = wave_sum(cc);
        if (t == 0) {
            float s0 = s_avg[0], c0 = c_avg[0];
            float ds2 = fmaxf(s0 * s0 - ss, EPSF);
            float dc2 = fmaxf(c0 * c0 - cc, EPSF);
            scal[0] = sqrtf(ds2);
            scal[1] = sqrtf(dc2);
            scal[2] = sqrtf(1.f + ss / ds2 + cc / dc2);
        }
    }
    __syncthreads();
    float* o = co + b * 201;
    if (t == 0) o[0] = scal[2];
    if (t >= 1 && t <= 100) o[t] = s_avg[t] / scal[0];
    if (t >= 101 && t <= 200) o[t] = c_avg[t - 100] / scal[1];
}

// ---------------------------------------------------------------------------
extern "C" void kernel_launch(void* const* d_in, const int* in_sizes, int n_in,
                              void* d_out, int out_size, void* d_ws, size_t ws_size,
                              hipStream_t stream) {
    (void)in_sizes; (void)n_in; (void)out_size; (void)ws_size;

    const float* Xs   = (const float*)d_in[0];   // (128,512,101)
    const float* Xc   = (const float*)d_in[1];   // (128,512,101)
    const float* W_l  = (const float*)d_in[2];   // (101,101)
    const float* b_l  = (const float*)d_in[3];
    const float* s_l  = (const float*)d_in[4];
    const float* W_s  = (const float*)d_in[5];   // (129,101)
    const float* b_s  = (const float*)d_in[6];
    const float* s_s  = (const float*)d_in[7];
    const float* W_c  = (const float*)d_in[8];   // (129,101)
    const float* b_c  = (const float*)d_in[9];
    const float* s_c  = (const float*)d_in[10];
    const float* w_hs = (const float*)d_in[11];  // (1,129)
    const float* w_hc = (const float*)d_in[12];  // (1,129)

    float* out = (float*)d_out;
    float* out_co = out;                 // 128*201
    float* out_as = out + 128 * 201;     // 128*512
    float* out_ac = out_as + 128 * 512;  // 128*512

    // workspace layout (floats)
    float* w = (float*)d_ws;
    size_t off = 0;
    auto alloc = [&](size_t n) { size_t o = off; off += n; return o; };
    const size_t NB = 128, NQ = 512;
    size_t oXs = alloc(NB * NQ * 104);
    size_t oXc = alloc(NB * NQ * 104);
    size_t oWl = alloc(112 * 104);
    size_t oWs = alloc(144 * 104);
    size_t oWc = alloc(144 * 104);
    size_t obl = alloc(112);
    size_t obs = alloc(144);
    size_t obc = alloc(144);
    size_t oSL  = alloc(NB * NQ * 112);  // sl
    size_t oWSb = alloc(NB * NQ * 144);  // ws_sent
    size_t oWCb = alloc(NB * NQ * 144);  // wc_comm
    size_t oC   = alloc(NB * NQ * NQ);   // C, then A_cs in place
    size_t oEsc = alloc(NB * NQ * NQ);   // A_sc transposed
    size_t oRM = alloc(NB * NQ), oRS = alloc(NB * NQ);
    size_t oCM = alloc(NB * NQ), oCS = alloc(NB * NQ);
    size_t oCtS = alloc(NB * NQ * 144);  // centroid for H_s (A_sc @ wc)
    size_t oCtC = alloc(NB * NQ * 144);  // centroid for H_c (A_cs @ ws)
    size_t oScS = alloc(NB * NQ);
    size_t oScC = alloc(NB * NQ);

    // 1) padding
    pad_x_kernel<<<dim3(512, 128), 128, 0, stream>>>(Xs, w + oXs, 0);
    pad_x_kernel<<<dim3(512, 128), 128, 0, stream>>>(Xc, w + oXc, 1);
    pad_w_kernel<<<112, 128, 0, stream>>>(W_l, b_l, w + oWl, w + obl, 101);
    pad_w_kernel<<<144, 128, 0, stream>>>(W_s, b_s, w + oWs, w + obs, 129);
    pad_w_kernel<<<144, 128, 0, stream>>>(W_c, b_c, w + oWc, w + obc, 129);

    // 2) Lorentz linears (WMMA GEMMs + bias), then rowwise fixup
    gemm_abt_nt<7><<<dim3(4, 128), 256, 0, stream>>>(
        w + oXs, 104, NQ * 104, w + oWl, 104, 0,
        w + oSL, 112, NQ * 112, w + obl, 1.f, 0.f, 1, 104);
    gemm_abt_nt<3><<<dim3(12, 128), 256, 0, stream>>>(
        w + oXs, 104, NQ * 104, w + oWs, 104, 0,
        w + oWSb, 144, NQ * 144, w + obs, 1.f, 0.f, 3, 104);
    gemm_abt_nt<3><<<dim3(12, 128), 256, 0, stream>>>(
        w + oXc, 104, NQ * 104, w + oWc, 104, 0,
        w + oWCb, 144, NQ * 144, w + obc, 1.f, 0.f, 3, 104);
    lorentz_fix_kernel<<<dim3(64, 128), 256, 0, stream>>>(w + oSL, 112, 101, s_l);
    lorentz_fix_kernel<<<dim3(64, 128), 256, 0, stream>>>(w + oWSb, 144, 129, s_s);
    lorentz_fix_kernel<<<dim3(64, 128), 256, 0, stream>>>(w + oWCb, 144, 129, s_c);

    // 3) C = 2 + 2 * (Xc_flip @ sl^T)   (WMMA GEMM, 16x64 macro-tiles)
    gemm_abt_nt<4><<<dim3(32, 128), 256, 0, stream>>>(
        w + oXc, 104, NQ * 104, w + oSL, 112, NQ * 112,
        w + oC, 512, NQ * NQ, nullptr, 2.f, 2.f, 8, 104);

    // 4) softmax stats (row and column)
    stats_kernel<<<dim3(64, 128, 2), 256, 0, stream>>>(
        w + oC, w + oRM, w + oRS, w + oCM, w + oCS);

    // 5) normalize: C := A_cs (in place), EscT := A_sc^T (LDS-tiled transpose)
    normexp_kernel<<<dim3(16, 16, 128), dim3(32, 8), 0, stream>>>(
        w + oC, w + oRM, w + oRS, w + oCM, w + oCS, w + oEsc);

    // 6) centroid GEMMs (pure WMMA, exp-free)
    gemm_ab_nt<3><<<dim3(12, 128), 256, 0, stream>>>(
        w + oEsc, 512, NQ * NQ, w + oWCb, 144, NQ * 144,
        w + oCtS, 144, NQ * 144, 3, 512);                  // A_sc @ wc_comm
    gemm_ab_nt<3><<<dim3(12, 128), 256, 0, stream>>>(
        w + oC, 512, NQ * NQ, w + oWSb, 144, NQ * 144,
        w + oCtC, 144, NQ * 144, 3, 512);                  // A_cs @ ws_sent

    // 7) Lorentz addition + tanh + project + attention score
    hs_score_kernel<<<dim3(64, 128), 256, 0, stream>>>(w + oCtS, w + oWSb, w_hs, w + oScS);
    hs_score_kernel<<<dim3(64, 128), 256, 0, stream>>>(w + oCtC, w + oWCb, w_hc, w + oScC);

    // 8) softmax over 512 -> a_s, a_c (straight into d_out)
    softmax512_kernel<<<dim3(128, 2), 32, 0, stream>>>(w + oScS, w + oScC, out_as, out_ac);

    // 9) final centroids + projected concat
    final_kernel<<<128, 256, 0, stream>>>(out_as, out_ac, Xs, Xc, out_co);
}